// BandedJointEncoder_44427141710186
// MI455X (gfx1250) — compile-verified
//
#include <hip/hip_runtime.h>
#include <math.h>

typedef __attribute__((ext_vector_type(2))) float v2f;
typedef __attribute__((ext_vector_type(4))) float v4f;
typedef __attribute__((ext_vector_type(8))) float v8f;

#define B_   4
#define T_   1024
#define D_   64
#define Z_   16
#define NCH  48
#define MEAN_ELEMS (B_ * Z_ * T_)   // 65536

__device__ __forceinline__ float softplus_f(float v) {
    // log(1+exp(v)), overflow-safe
    return (v > 20.0f) ? v : log1pf(expf(v));
}

// ---------------------------------------------------------------------------
// Kernel A: fp32 WMMA GEMM (4096x64 @ 64x48) + head transform.
// One wave per 16x16 output tile; K-loop of 16 steps of V_WMMA_F32_16X16X4_F32.
//   ntile 0       -> mapped_mean (transposed to (B,16,T))
//   ntile 1,2     -> softplus; even channel: d=sp+1 -> invd; odd: s;
//                    pair via __shfl_xor(1); store invd, t=-s/d to workspace.
// ---------------------------------------------------------------------------
__global__ __launch_bounds__(32) void bje_gemm_head(
    const float* __restrict__ x, const float* __restrict__ W,
    const float* __restrict__ bias,
    float* __restrict__ out_mean,
    float* __restrict__ ws_t, float* __restrict__ ws_invd)
{
    const int wtile = blockIdx.x;        // 0..767  (256 mtiles x 3 ntiles)
    const int mtile = wtile / 3;
    const int ntile = wtile % 3;
    const int lane  = threadIdx.x;       // wave32
    const int n     = lane & 15;         // column within tile / A row
    const int kh    = lane >> 4;         // K-half select (0 or 1)

    v8f acc = {};
    #pragma unroll
    for (int k = 0; k < D_; k += 4) {
        // A fragment 16x4: lane m holds (K=2*kh, K=2*kh+1) of row m
        v2f a;
        const float* ap = x + (size_t)(mtile * 16 + n) * D_ + k + 2 * kh;
        a.x = ap[0];
        a.y = ap[1];
        // B fragment 4x16: lane n holds rows K=2*kh, 2*kh+1 of column n
        v2f bm;
        const float* bp = W + (size_t)(k + 2 * kh) * NCH + ntile * 16 + n;
        bm.x = bp[0];
        bm.y = bp[NCH];
        acc = __builtin_amdgcn_wmma_f32_16x16x4_f32(
            /*neg_a=*/false, a, /*neg_b=*/false, bm,
            /*c_mod=*/(short)0, acc, /*reuse_a=*/false, /*reuse_b=*/false);
    }

    const int ch = ntile * 16 + n;       // output channel 0..47
    const float bv = bias[ch];
    #pragma unroll
    for (int j = 0; j < 8; ++j) acc[j] += bv;

    if (ntile == 0) {
        // mean output, transposed layout (B, 16, T)
        #pragma unroll
        for (int j = 0; j < 8; ++j) {
            const int g    = mtile * 16 + j + 8 * kh;  // global x row
            const int bi   = g >> 10;
            const int trow = g & (T_ - 1);
            out_mean[((size_t)(bi * Z_ + ch)) * T_ + trow] = acc[j];
        }
    } else {
        const int cc     = ch - 16;      // 0..31
        const int z      = cc >> 1;
        const int parity = cc & 1;
        #pragma unroll
        for (int j = 0; j < 8; ++j) {
            const float sp    = softplus_f(acc[j]);
            const float other = __shfl_xor(sp, 1, 32);  // partner channel's softplus
            if (parity == 0) {
                const int g    = mtile * 16 + j + 8 * kh;
                const int bi   = g >> 10;
                const int trow = g & (T_ - 1);
                const float dg   = sp + 1.0f;       // diag
                const float invd = 1.0f / dg;
                const size_t idx = ((size_t)(bi * Z_ + z)) * T_ + trow;
                ws_invd[idx] = invd;
                ws_t[idx]    = -other * invd;       // t_k = -s_k / d_k
            }
        }
    }
}

// ---------------------------------------------------------------------------
// Kernel B: materialize scale_tril. Store-bandwidth-bound (268 MB out).
// L[r,c] = invd_r * prod_{k=c}^{r-1} t_k  (c<=r), else 0.
// Block: 256 threads (8 waves), handles 64 rows of one (b,z); t/invd staged
// once in LDS. Per row, wave sweeps 8 chunks of 128 cols right-to-left:
// per-lane float4 suffix products + shuffle product-scan + running carry.
// Non-temporal b128 stores (streaming output, keep L2 clean).
// ---------------------------------------------------------------------------
__global__ __launch_bounds__(256) void bje_bidiag_inv(
    const float* __restrict__ ws_t, const float* __restrict__ ws_invd,
    float* __restrict__ out_tril)
{
    __shared__ float t_s[T_];
    __shared__ float id_s[T_];

    const int zb  = blockIdx.y;          // 0..63 = b*16+z
    const int tid = threadIdx.x;
    for (int i = tid; i < T_; i += 256) {
        t_s[i]  = ws_t[(size_t)zb * T_ + i];
        id_s[i] = ws_invd[(size_t)zb * T_ + i];
    }
    __syncthreads();

    const int wave = tid >> 5;
    const int lane = tid & 31;
    float* base = out_tril + (size_t)zb * T_ * T_;

    #pragma unroll 1
    for (int rr = 0; rr < 8; ++rr) {
        const int   r    = blockIdx.x * 64 + wave * 8 + rr;
        const float invd = id_s[r];
        float* rowp = base + (size_t)r * T_;
        float carry = 1.0f;                       // prod of t over chunks right of cb

        #pragma unroll 1
        for (int cb = T_ - 128; cb >= 0; cb -= 128) {
            const int c0 = cb + lane * 4;
            const v4f tt = *reinterpret_cast<const v4f*>(&t_s[c0]);
            // mask: factors only for k < r
            const float m0 = (c0 + 0 < r) ? tt.x : 1.0f;
            const float m1 = (c0 + 1 < r) ? tt.y : 1.0f;
            const float m2 = (c0 + 2 < r) ? tt.z : 1.0f;
            const float m3 = (c0 + 3 < r) ? tt.w : 1.0f;
            // per-lane suffix products
            const float q3 = m3;
            const float q2 = m2 * q3;
            const float q1 = m1 * q2;
            const float q0 = m0 * q1;
            // cross-lane reverse inclusive product scan: I_l = prod_{j>=l} p_j
            float I = q0;
            #pragma unroll
            for (int off = 1; off < 32; off <<= 1) {
                const float o = __shfl_down(I, off, 32);
                if (lane + off < 32) I *= o;
            }
            const float chunkprod = __shfl(I, 0, 32);       // whole-chunk product
            float E = __shfl_down(I, 1, 32);                // exclusive (lanes > l)
            if (lane == 31) E = 1.0f;

            const float sbase = E * carry * invd;
            v4f o4;
            o4.x = (c0 + 0 <= r) ? q0 * sbase : 0.0f;
            o4.y = (c0 + 1 <= r) ? q1 * sbase : 0.0f;
            o4.z = (c0 + 2 <= r) ? q2 * sbase : 0.0f;
            o4.w = (c0 + 3 <= r) ? q3 * sbase : 0.0f;
            __builtin_nontemporal_store(o4, reinterpret_cast<v4f*>(rowp + c0));

            carry *= chunkprod;
        }
    }
}

extern "C" void kernel_launch(void* const* d_in, const int* in_sizes, int n_in,
                              void* d_out, int out_size, void* d_ws, size_t ws_size,
                              hipStream_t stream) {
    const float* x    = (const float*)d_in[0];   // (4,1024,64)
    const float* W    = (const float*)d_in[1];   // (64,48)
    const float* bias = (const float*)d_in[2];   // (48,)

    float* out      = (float*)d_out;
    float* out_mean = out;                       // (4,16,1024)
    float* out_tril = out + MEAN_ELEMS;          // (4,16,1024,1024)

    float* ws_t    = (float*)d_ws;               // 64*1024 floats
    float* ws_invd = ws_t + (size_t)B_ * Z_ * T_;

    bje_gemm_head<<<dim3(768), dim3(32), 0, stream>>>(x, W, bias,
                                                      out_mean, ws_t, ws_invd);
    bje_bidiag_inv<<<dim3(T_ / 64, B_ * Z_), dim3(256), 0, stream>>>(ws_t, ws_invd,
                                                                     out_tril);
}